// PredCodingNet_47390669144433
// MI455X (gfx1250) — compile-verified
//
#include <hip/hip_runtime.h>

// CDNA5 (gfx1250) fp32 WMMA: D(16x16) = A(16x4) * B(4x16) + C, wave32.
typedef __attribute__((ext_vector_type(2))) float v2f;
typedef __attribute__((ext_vector_type(8))) float v8f;

#define WAVES 8   // 256 threads per block = 8 wave32

// C = relu(A @ W^T), A:[M,K] row-major, W:[Nvalid,K] row-major, C:[M,Nout].
// One wave computes a (16*MTILE) x (16*NTILE) output strip with register
// double-buffering of the A/B fragments so WMMAs overlap in-flight loads.
template <int MTILE, int NTILE>
__global__ __launch_bounds__(WAVES * 32)
void gemm_nt_relu(const float* __restrict__ A,
                  const float* __restrict__ W,
                  float* __restrict__ C,
                  int K, int Nout, int Nvalid)
{
    const int lane = threadIdx.x & 31;
    const int wave = threadIdx.x >> 5;
    const int l16  = lane & 15;   // row/col within a 16x16 tile
    const int hi   = lane >> 4;   // lane half: selects K pair / M half
    const int koff = hi * 2;      // K offset inside each 4-wide K step

    const int mBase = (blockIdx.x * WAVES + wave) * (16 * MTILE);
    const int nBase = blockIdx.y * (16 * NTILE);

    // A fragments: lane holds row (mBase+i*16+l16), contiguous K pair.
    const float* aPtr[MTILE];
#pragma unroll
    for (int i = 0; i < MTILE; ++i)
        aPtr[i] = A + (size_t)(mBase + i * 16 + l16) * K + koff;

    // B = W^T fragments: lane holds W row (nBase+j*16+l16), same K-pair pattern.
    const float* wPtr[NTILE];
    int nCol[NTILE];
#pragma unroll
    for (int j = 0; j < NTILE; ++j) {
        nCol[j] = nBase + j * 16 + l16;
        int nc = nCol[j] < Nvalid ? nCol[j] : (Nvalid - 1);  // clamp (GEMM3 pad)
        wPtr[j] = W + (size_t)nc * K + koff;
    }

    v8f acc[MTILE][NTILE] = {};

    // ---- software pipeline: prologue loads for k = 0 ----
    v2f a_c[MTILE], b_c[NTILE];
#pragma unroll
    for (int i = 0; i < MTILE; ++i) a_c[i] = *(const v2f*)(aPtr[i]);
#pragma unroll
    for (int j = 0; j < NTILE; ++j) b_c[j] = *(const v2f*)(wPtr[j]);

    int k = 0;
#pragma unroll 2
    for (; k + 4 < K; k += 4) {
        // issue next k-step's loads first (stay in flight during the WMMAs)
        v2f a_n[MTILE], b_n[NTILE];
#pragma unroll
        for (int i = 0; i < MTILE; ++i) a_n[i] = *(const v2f*)(aPtr[i] + k + 4);
#pragma unroll
        for (int j = 0; j < NTILE; ++j) b_n[j] = *(const v2f*)(wPtr[j] + k + 4);

        // pull the streaming A operand toward the WGP ahead of time
        if (k + 132 < K) {
#pragma unroll
            for (int i = 0; i < MTILE; ++i)
                __builtin_prefetch(aPtr[i] + k + 132, 0, 0);
        }

        // compute on the already-resident fragments
#pragma unroll
        for (int i = 0; i < MTILE; ++i)
#pragma unroll
            for (int j = 0; j < NTILE; ++j)
                acc[i][j] = __builtin_amdgcn_wmma_f32_16x16x4_f32(
                    false, a_c[i], false, b_c[j], (short)0, acc[i][j], false, false);

        // rotate buffers
#pragma unroll
        for (int i = 0; i < MTILE; ++i) a_c[i] = a_n[i];
#pragma unroll
        for (int j = 0; j < NTILE; ++j) b_c[j] = b_n[j];
    }

    // ---- epilogue k-step ----
#pragma unroll
    for (int i = 0; i < MTILE; ++i)
#pragma unroll
        for (int j = 0; j < NTILE; ++j)
            acc[i][j] = __builtin_amdgcn_wmma_f32_16x16x4_f32(
                false, a_c[i], false, b_c[j], (short)0, acc[i][j], false, false);

    // C/D layout: VGPR v, lane-half hi -> M = base + 8*hi + v, N = nCol.
#pragma unroll
    for (int i = 0; i < MTILE; ++i) {
        const int mRow = mBase + i * 16 + hi * 8;
#pragma unroll
        for (int j = 0; j < NTILE; ++j) {
            if (nCol[j] < Nvalid) {
#pragma unroll
                for (int v = 0; v < 8; ++v) {
                    float r = acc[i][j][v];
                    r = r > 0.0f ? r : 0.0f;               // ReLU epilogue
                    C[(size_t)(mRow + v) * Nout + nCol[j]] = r;
                }
            }
        }
    }
}

extern "C" void kernel_launch(void* const* d_in, const int* in_sizes, int n_in,
                              void* d_out, int out_size, void* d_ws, size_t ws_size,
                              hipStream_t stream) {
    // setup_inputs order: x, W0, W1, W2, b0, b1, b2.
    // Biases are unused in the forward path, and the 15-iteration PC loop never
    // affects m3 (the returned tensor), so only the 3 forward GEMMs are needed.
    const float* x  = (const float*)d_in[0];
    const float* W0 = (const float*)d_in[1];   // [512, 784]
    const float* W1 = (const float*)d_in[2];   // [256, 512]
    const float* W2 = (const float*)d_in[3];   // [ 10, 256]
    float* out = (float*)d_out;                // [32768, 10]

    const int M = 32768;
    float* m1 = (float*)d_ws;                  // [32768, 512] fp32 (64 MB, L2-resident)
    float* m2 = m1 + (size_t)M * 512;          // [32768, 256] fp32 (32 MB)

    dim3 blk(WAVES * 32);
    const int mBlk = WAVES * 16 * 2;           // 256 rows per block (MTILE=2)

    // m1 = relu(x @ W0^T)   : K=784, N=512
    gemm_nt_relu<2, 4><<<dim3(M / mBlk, 512 / 64), blk, 0, stream>>>(
        x, W0, m1, 784, 512, 512);

    // m2 = relu(m1 @ W1^T)  : K=512, N=256
    gemm_nt_relu<2, 4><<<dim3(M / mBlk, 256 / 64), blk, 0, stream>>>(
        m1, W1, m2, 512, 256, 256);

    // out = relu(m2 @ W2^T) : K=256, N=10 (padded to one 16-wide tile)
    gemm_nt_relu<2, 1><<<dim3(M / mBlk, 1), blk, 0, stream>>>(
        m2, W2, out, 256, 10, 10);
}